// FeatureEncoder_11115375362797
// MI455X (gfx1250) — compile-verified
//
#include <hip/hip_runtime.h>
#include <stdint.h>

#define B_     8
#define N_     4096
#define S_     1024
#define KNN_   32
#define PTCH_  64            // IN_CH - 3
#define NT_    (B_*S_*KNN_)  // 262144 grouped samples
#define NT2_   (B_*S_)       // 8192 centers
#define EPS_   1e-5f
#define FPS_PPT 4            // N_/1024 points per thread in FPS

typedef float v2f __attribute__((ext_vector_type(2)));
typedef float v8f __attribute__((ext_vector_type(8)));

// ---------------- wave32 cross-lane helpers ----------------
__device__ __forceinline__ void wave_argmax(float& v, int& i) {
#pragma unroll
  for (int o = 16; o > 0; o >>= 1) {
    float ov = __shfl_xor(v, o, 32);
    int   oi = __shfl_xor(i, o, 32);
    if (ov > v || (ov == v && oi < i)) { v = ov; i = oi; }
  }
}
__device__ __forceinline__ void wave_argmin(float& v, int& i) {
#pragma unroll
  for (int o = 16; o > 0; o >>= 1) {
    float ov = __shfl_xor(v, o, 32);
    int   oi = __shfl_xor(i, o, 32);
    if (ov < v || (ov == v && oi < i)) { v = ov; i = oi; }
  }
}
__device__ __forceinline__ float wave_max(float v) {
#pragma unroll
  for (int o = 16; o > 0; o >>= 1) v = fmaxf(v, __shfl_xor(v, o, 32));
  return v;
}

// ---------------- CDNA5 async global->LDS staging ----------------
// LDS byte offset lives in addr[31:0] of the flat address of a __shared__ object.
__device__ __forceinline__ void async_g2l_b32(void* ldsdst, const float* gsrc) {
  uint32_t d = (uint32_t)(size_t)ldsdst;
  asm volatile("global_load_async_to_lds_b32 %0, %1, off"
               :: "v"(d), "v"((uint64_t)(size_t)gsrc)
               : "memory");
}
__device__ __forceinline__ void wait_async0() {
  asm volatile("s_wait_asynccnt 0" ::: "memory");
}

// ---------------- 1) Gaussian KDE inverse density ----------------
// xyz (B,3,N) -> inv_dens (B,N). All-pairs, xyz tile staged via async LDS DMA.
__global__ void k_density(const float* __restrict__ xyz, float* __restrict__ inv_dens) {
  const int b = blockIdx.y;
  const int i = blockIdx.x * 256 + threadIdx.x;
  const float* X = xyz + (size_t)b * 3 * N_;
  const float xi = X[i], yi = X[N_ + i], zi = X[2 * N_ + i];
  __shared__ float sx[256], sy[256], sz[256];
  float acc = 0.f;
  for (int j0 = 0; j0 < N_; j0 += 256) {
    async_g2l_b32(&sx[threadIdx.x], X + j0 + threadIdx.x);
    async_g2l_b32(&sy[threadIdx.x], X + N_ + j0 + threadIdx.x);
    async_g2l_b32(&sz[threadIdx.x], X + 2 * N_ + j0 + threadIdx.x);
    wait_async0();
    __syncthreads();
#pragma unroll 8
    for (int j = 0; j < 256; ++j) {
      float dx = xi - sx[j], dy = yi - sy[j], dz = zi - sz[j];
      float d2 = dx * dx + dy * dy + dz * dz;
      acc += __expf(-2.0f * d2);             // exp(-d2 / (2*0.5^2))
    }
    __syncthreads();
  }
  float dens = acc * (1.0f / (1.25f * (float)N_));  // / (2.5*BW) then mean over N
  inv_dens[(size_t)b * N_ + i] = 1.0f / dens;
}

// ---------------- 2) Farthest point sampling ----------------
__global__ void k_fps(const float* __restrict__ xyz, int* __restrict__ fps_idx,
                      float* __restrict__ new_xyz) {
  const int b = blockIdx.x, t = threadIdx.x;
  const int wave = t >> 5, lane = t & 31;
  const float* X = xyz + (size_t)b * 3 * N_;
  float px[FPS_PPT], py[FPS_PPT], pz[FPS_PPT], dist[FPS_PPT];
#pragma unroll
  for (int p = 0; p < FPS_PPT; ++p) {
    int i = p * 1024 + t;
    px[p] = X[i]; py[p] = X[N_ + i]; pz[p] = X[2 * N_ + i];
    dist[p] = 1e10f;
  }
  __shared__ float rv[32];
  __shared__ int   ri[32];
  __shared__ int   s_far;
  if (t == 0) s_far = 0;
  __syncthreads();
  for (int s = 0; s < S_; ++s) {
    const int far = s_far;
    const float cx = X[far], cy = X[N_ + far], cz = X[2 * N_ + far];
    if (t == 0) {
      fps_idx[b * S_ + s] = far;
      new_xyz[(size_t)(b * S_ + s) * 3 + 0] = cx;
      new_xyz[(size_t)(b * S_ + s) * 3 + 1] = cy;
      new_xyz[(size_t)(b * S_ + s) * 3 + 2] = cz;
    }
    float bv = -1.f; int bi = 0;
#pragma unroll
    for (int p = 0; p < FPS_PPT; ++p) {
      float dx = px[p] - cx, dy = py[p] - cy, dz = pz[p] - cz;
      float d2 = dx * dx + dy * dy + dz * dz;
      dist[p] = fminf(dist[p], d2);
      int i = p * 1024 + t;
      if (dist[p] > bv || (dist[p] == bv && i < bi)) { bv = dist[p]; bi = i; }
    }
    wave_argmax(bv, bi);
    if (lane == 0) { rv[wave] = bv; ri[wave] = bi; }
    __syncthreads();
    if (wave == 0) {
      float v = rv[lane]; int i = ri[lane];
      wave_argmax(v, i);
      if (lane == 0) s_far = i;
    }
    __syncthreads();
  }
}

// ---------------- 3) kNN (top-32 smallest d2 of 4096) ----------------
__global__ void k_knn(const float* __restrict__ xyz, const float* __restrict__ new_xyz,
                      int* __restrict__ knn) {
  extern __shared__ float sm[];
  const int wave = threadIdx.x >> 5, lane = threadIdx.x & 31;
  const int c = blockIdx.x * 4 + wave;
  const int b = c / S_;
  float* d = sm + (size_t)wave * N_;
  const float cx = new_xyz[(size_t)c * 3 + 0];
  const float cy = new_xyz[(size_t)c * 3 + 1];
  const float cz = new_xyz[(size_t)c * 3 + 2];
  const float* X = xyz + (size_t)b * 3 * N_;
  for (int j = lane; j < N_; j += 32) {
    float dx = X[j] - cx, dy = X[N_ + j] - cy, dz = X[2 * N_ + j] - cz;
    d[j] = dx * dx + dy * dy + dz * dz;
  }
  __syncthreads();
  for (int p = 0; p < KNN_; ++p) {
    float bv = 3.4e38f; int bi = 0x7fffffff;
    for (int j = lane; j < N_; j += 32) {
      float v = d[j];
      if (v < bv || (v == bv && j < bi)) { bv = v; bi = j; }
    }
    wave_argmin(bv, bi);
    if (lane == 0) { knn[(size_t)c * KNN_ + p] = bi; d[bi] = 3.4e38f; }
    __syncthreads();
  }
}

// ---------------- 4) Grouping: build GEMM-ready feature planes ----------------
__global__ void k_group(const float* __restrict__ xyz, const float* __restrict__ points,
                        const float* __restrict__ new_xyz, const float* __restrict__ inv_dens,
                        const int* __restrict__ knn,
                        float* __restrict__ X0, float* __restrict__ W0, float* __restrict__ D0) {
  const int wave = threadIdx.x >> 5, lane = threadIdx.x & 31;
  const int c = blockIdx.x * 4 + wave;
  const int b = c / S_;
  const int j = knn[(size_t)c * KNN_ + lane];
  const size_t n = (size_t)c * KNN_ + lane;
  const float* X = xyz + (size_t)b * 3 * N_;
  const float gx = X[j]          - new_xyz[(size_t)c * 3 + 0];
  const float gy = X[N_ + j]     - new_xyz[(size_t)c * 3 + 1];
  const float gz = X[2 * N_ + j] - new_xyz[(size_t)c * 3 + 2];
  X0[0 * (size_t)NT_ + n] = gx;
  X0[1 * (size_t)NT_ + n] = gy;
  X0[2 * (size_t)NT_ + n] = gz;
  const float* P = points + (size_t)b * PTCH_ * N_;
#pragma unroll 4
  for (int ci = 0; ci < PTCH_; ++ci)
    X0[(size_t)(3 + ci) * NT_ + n] = P[(size_t)ci * N_ + j];
  X0[(size_t)67 * NT_ + n] = 0.f;   // K-pad row
  W0[0 * (size_t)NT_ + n] = gx;
  W0[1 * (size_t)NT_ + n] = gy;
  W0[2 * (size_t)NT_ + n] = gz;
  W0[3 * (size_t)NT_ + n] = 0.f;
  const float gd = inv_dens[(size_t)b * N_ + j];
  const float m = wave_max(gd);     // max over K within the (b,s) group
  D0[0 * (size_t)NT_ + n] = gd / m;
  D0[1 * (size_t)NT_ + n] = 0.f;
  D0[2 * (size_t)NT_ + n] = 0.f;
  D0[3 * (size_t)NT_ + n] = 0.f;
}

// ---------------- 5a) zero-pad weights into Mpad x Kpad tile ----------------
__global__ void k_padw(const float* __restrict__ W, float* __restrict__ Wp,
                       int Cout, int Cin, int Kpad, int nel) {
  const int idx = blockIdx.x * 256 + threadIdx.x;
  if (idx >= nel) return;
  const int m = idx / Kpad, k = idx - m * Kpad;
  Wp[idx] = (m < Cout && k < Cin) ? W[m * Cin + k] : 0.f;
}

// ---------------- 5b) fp32 WMMA GEMM: Y(Mpad x NT) = Wp(Mpad x Kpad) * X(Kpad x NT) --------
// Branch-free: weights pre-padded, K a compile-time constant -> fully unrolled
// WMMA chain for conv layers, 4x-unrolled + prefetch for the 2048-deep linear.
template <int KPAD, int NTN>
__global__ void k_gemm(const float* __restrict__ Wp, const float* __restrict__ Xm,
                       float* __restrict__ Ym) {
  const int wave = threadIdx.x >> 5, lane = threadIdx.x & 31;
  const int n0 = (blockIdx.x * 8 + wave) * 16;
  const int m0 = blockIdx.y * 16;
  const int mrow = lane & 15, kh = lane >> 4;
  const int nB = n0 + mrow;
  const float* wrow = Wp + (size_t)(m0 + mrow) * KPAD + 2 * kh;   // A: contiguous pair
  const float* xcol = Xm + (size_t)(2 * kh) * NTN + nB;           // B: two rows apart
  v8f acc = {0.f, 0.f, 0.f, 0.f, 0.f, 0.f, 0.f, 0.f};
  if constexpr (KPAD <= 128) {
#pragma unroll
    for (int k0 = 0; k0 < KPAD; k0 += 4) {
      v2f a = *(const v2f*)(wrow + k0);
      v2f b;
      b.x = xcol[(size_t)k0 * NTN];
      b.y = xcol[(size_t)(k0 + 1) * NTN];
      acc = __builtin_amdgcn_wmma_f32_16x16x4_f32(false, a, false, b, (short)0, acc,
                                                  false, false);
    }
  } else {
#pragma unroll 4
    for (int k0 = 0; k0 < KPAD; k0 += 4) {
      v2f a = *(const v2f*)(wrow + k0);
      v2f b;
      b.x = xcol[(size_t)k0 * NTN];
      b.y = xcol[(size_t)(k0 + 1) * NTN];
      if (k0 + 4 < KPAD)
        __builtin_prefetch(xcol + (size_t)(k0 + 4) * NTN, 0, 1);
      acc = __builtin_amdgcn_wmma_f32_16x16x4_f32(false, a, false, b, (short)0, acc,
                                                  false, false);
    }
  }
#pragma unroll
  for (int v = 0; v < 8; ++v)
    Ym[(size_t)(m0 + 8 * kh + v) * NTN + nB] = acc[v];
}

// ---------------- 6) BN stats (sum, sumsq per channel) ----------------
__global__ void k_zero(float* __restrict__ p, int nel) {
  if ((int)threadIdx.x < nel) p[threadIdx.x] = 0.f;
}

__global__ void k_colstats(const float* __restrict__ Ym, float* __restrict__ stats, int NTn) {
  const int c = blockIdx.y;
  const int t = threadIdx.x;
  float a = 0.f, b2 = 0.f;
  for (int n = blockIdx.x * 256 + t; n < NTn; n += gridDim.x * 256) {
    float v = Ym[(size_t)c * NTn + n];
    a += v; b2 += v * v;
  }
  __shared__ float sa[256], sb[256];
  sa[t] = a; sb[t] = b2;
  __syncthreads();
  for (int o = 128; o > 0; o >>= 1) {
    if (t < o) { sa[t] += sa[t + o]; sb[t] += sb[t + o]; }
    __syncthreads();
  }
  if (t == 0) {
    atomicAdd(&stats[2 * c],     sa[0]);
    atomicAdd(&stats[2 * c + 1], sb[0]);
  }
}

// ---------------- 7) BN + ReLU (+ zero K-pad rows for next layer) ----------------
__global__ void k_bn_relu(const float* __restrict__ Ym, const float* __restrict__ stats,
                          const float* __restrict__ g, const float* __restrict__ be,
                          float* __restrict__ Om, int Cout, int NTn) {
  const int c = blockIdx.y;
  const int n = blockIdx.x * 256 + threadIdx.x;
  float o = 0.f;
  if (c < Cout) {
    const float inv = 1.f / (float)NTn;
    const float m   = stats[2 * c] * inv;
    const float var = stats[2 * c + 1] * inv - m * m;
    const float r   = rsqrtf(var + EPS_);
    const float v   = (Ym[(size_t)c * NTn + n] - m) * r * g[c] + be[c];
    o = fmaxf(v, 0.f);
  }
  Om[(size_t)c * NTn + n] = o;
}

// ---------------- 8) x *= density scale (broadcast over 128 channels) ----------------
__global__ void k_mult(float* __restrict__ X3, const float* __restrict__ dsf) {
  const int c = blockIdx.y;
  const size_t n = (size_t)blockIdx.x * 256 + threadIdx.x;
  X3[(size_t)c * NT_ + n] *= dsf[n];
}

// ---------------- 9) per-center (128x32)x(32x16) aggregation via WMMA ----------------
// wave = one (center, 16-channel subtile); both fragments load as b64 pairs.
__global__ void k_agg(const float* __restrict__ X3, const float* __restrict__ W3,
                      float* __restrict__ aggT) {
  const int bs = blockIdx.x;
  const int ct = threadIdx.x >> 5, lane = threadIdx.x & 31;
  const int mrow = lane & 15, kh = lane >> 4;
  const size_t nb = (size_t)bs * KNN_;
  const float* arow = X3 + (size_t)(ct * 16 + mrow) * NT_ + nb + 2 * kh;
  const float* brow = W3 + (size_t)mrow * NT_ + nb + 2 * kh;
  v8f acc = {0.f, 0.f, 0.f, 0.f, 0.f, 0.f, 0.f, 0.f};
#pragma unroll
  for (int k0 = 0; k0 < KNN_; k0 += 4) {
    v2f a = *(const v2f*)(arow + k0);
    v2f b = *(const v2f*)(brow + k0);
    acc = __builtin_amdgcn_wmma_f32_16x16x4_f32(false, a, false, b, (short)0, acc,
                                                false, false);
  }
#pragma unroll
  for (int v = 0; v < 8; ++v) {
    const int cch = ct * 16 + 8 * kh + v;
    aggT[(size_t)(cch * 16 + mrow) * NT2_ + bs] = acc[v];
  }
}

// ---------------- 10) final BN -> d_out in (B,128,S) layout ----------------
__global__ void k_bn_out(const float* __restrict__ Ym, const float* __restrict__ stats,
                         const float* __restrict__ g, const float* __restrict__ be,
                         float* __restrict__ out) {
  const int c = blockIdx.y;
  const int n = blockIdx.x * 256 + threadIdx.x;      // 0..8191 (b*S+s)
  const float inv = 1.f / (float)NT2_;
  const float m   = stats[2 * c] * inv;
  const float var = stats[2 * c + 1] * inv - m * m;
  const float r   = rsqrtf(var + EPS_);
  float v = (Ym[(size_t)c * NT2_ + n] - m) * r * g[c] + be[c];
  v = fmaxf(v, 0.f);
  const int b = n >> 10, s = n & 1023;
  out[(size_t)B_ * 3 * S_ + ((size_t)b * 128 + c) * S_ + s] = v;
}

// ---------------- 11) new_xyz -> d_out in (B,3,S) layout ----------------
__global__ void k_out_xyz(const float* __restrict__ nx, float* __restrict__ out) {
  const int n = blockIdx.x * 256 + threadIdx.x;      // b*S+s
  const int b = n >> 10, s = n & 1023;
#pragma unroll
  for (int d = 0; d < 3; ++d)
    out[((size_t)b * 3 + d) * S_ + s] = nx[(size_t)n * 3 + d];
}

// ---------------- conv + training-BN + ReLU layer driver ----------------
template <int KPAD, int NTN>
static void conv_bn(const float* W, const float* g, const float* be,
                    const float* Xin, float* Yraw, float* Out,
                    int Cout, int Cin, int Mpad,
                    float* Wp, float* stats, hipStream_t stream) {
  const int nel = Mpad * KPAD;
  k_padw<<<dim3((nel + 255) / 256), 256, 0, stream>>>(W, Wp, Cout, Cin, KPAD, nel);
  k_zero<<<1, 256, 0, stream>>>(stats, 256);
  k_gemm<KPAD, NTN><<<dim3(NTN / 128, Mpad / 16), 256, 0, stream>>>(Wp, Xin, Yraw);
  k_colstats<<<dim3(32, Cout), 256, 0, stream>>>(Yraw, stats, NTN);
  const int Cpad = (Cout + 3) & ~3;
  k_bn_relu<<<dim3(NTN / 256, Cpad), 256, 0, stream>>>(Yraw, stats, g, be, Out, Cout, NTN);
}

// =====================================================================
extern "C" void kernel_launch(void* const* d_in, const int* in_sizes, int n_in,
                              void* d_out, int out_size, void* d_ws, size_t ws_size,
                              hipStream_t stream) {
  (void)in_sizes; (void)n_in; (void)out_size; (void)ws_size;
  // JAX pytree flatten (sorted keys): xyz, points, bnl_b, bnl_g,
  // dn[i]{b,be,g,w}@4, lin_b@16, lin_w@17, mlp[i]{b,be,g,w}@18, wn[i]{b,be,g,w}@30
  const float* xyz    = (const float*)d_in[0];
  const float* points = (const float*)d_in[1];
  const float* bnl_b  = (const float*)d_in[2];
  const float* bnl_g  = (const float*)d_in[3];
  const float* lin_w  = (const float*)d_in[17];
  auto Lp = [&](int base, int i, int f) { return (const float*)d_in[base + 4 * i + f]; };

  float* ws = (float*)d_ws;
  size_t off = 0;
  float* invd   = ws + off; off += (size_t)B_ * N_;
  int*   fpsidx = (int*)(ws + off); off += NT2_;
  float* nxyz   = ws + off; off += (size_t)NT2_ * 3;
  int*   knn    = (int*)(ws + off); off += (size_t)NT2_ * KNN_;
  float* stats  = ws + off; off += 256;
  float* wpad   = ws + off; off += (size_t)128 * 2048;  // padded weight tile (max = lin_w)
  float* bufA   = ws + off; off += (size_t)128 * NT_;   // X0/X1/X2/X3
  float* bufB   = ws + off; off += (size_t)128 * NT_;   // GEMM raw / aggT
  float* wbA    = ws + off; off += (size_t)16 * NT_;    // weight-net ping
  float* wbB    = ws + off; off += (size_t)16 * NT_;    // weight-net pong
  float* dbA    = ws + off; off += (size_t)16 * NT_;    // density-net ping
  float* dbB    = ws + off; off += (size_t)16 * NT_;    // density-net pong
  float* ybuf   = ws + off; off += (size_t)128 * NT2_;  // final linear out
  float* out    = (float*)d_out;

  // Phase 1-4: density, FPS, kNN, grouping
  k_density<<<dim3(N_ / 256, B_), 256, 0, stream>>>(xyz, invd);
  k_fps<<<dim3(B_), 1024, 0, stream>>>(xyz, fpsidx, nxyz);
  k_knn<<<dim3(NT2_ / 4), 128, 4 * N_ * sizeof(float), stream>>>(xyz, nxyz, knn);
  k_group<<<dim3(NT2_ / 4), 128, 0, stream>>>(xyz, points, nxyz, invd, knn, bufA, wbA, dbA);

  // point MLP: 67 -> 64 -> 64 -> 128 (X chain ping-pongs bufA <-> bufB)
  conv_bn<68, NT_>(Lp(18, 0, 3), Lp(18, 0, 2), Lp(18, 0, 1), bufA, bufB, bufA,
                   64, 67, 64, wpad, stats, stream);
  conv_bn<64, NT_>(Lp(18, 1, 3), Lp(18, 1, 2), Lp(18, 1, 1), bufA, bufB, bufA,
                   64, 64, 64, wpad, stats, stream);
  conv_bn<64, NT_>(Lp(18, 2, 3), Lp(18, 2, 2), Lp(18, 2, 1), bufA, bufB, bufA,
                   128, 64, 128, wpad, stats, stream);
  // weight net: 3 -> 8 -> 8 -> 16 (final in wbA)
  conv_bn<4,  NT_>(Lp(30, 0, 3), Lp(30, 0, 2), Lp(30, 0, 1), wbA, wbB, wbA,
                   8, 3, 16, wpad, stats, stream);
  conv_bn<8,  NT_>(Lp(30, 1, 3), Lp(30, 1, 2), Lp(30, 1, 1), wbA, wbB, wbA,
                   8, 8, 16, wpad, stats, stream);
  conv_bn<8,  NT_>(Lp(30, 2, 3), Lp(30, 2, 2), Lp(30, 2, 1), wbA, wbB, wbA,
                   16, 8, 16, wpad, stats, stream);
  // density net: 1 -> 16 -> 8 -> 1 (all-ReLU per faithful reference; final in dbA row 0)
  conv_bn<4,  NT_>(Lp(4, 0, 3), Lp(4, 0, 2), Lp(4, 0, 1), dbA, dbB, dbA,
                   16, 1, 16, wpad, stats, stream);
  conv_bn<16, NT_>(Lp(4, 1, 3), Lp(4, 1, 2), Lp(4, 1, 1), dbA, dbB, dbA,
                   8, 16, 16, wpad, stats, stream);
  conv_bn<8,  NT_>(Lp(4, 2, 3), Lp(4, 2, 2), Lp(4, 2, 1), dbA, dbB, dbA,
                   1, 8, 16, wpad, stats, stream);

  // x *= density-scale, then per-center aggregation into aggT (bufB)
  k_mult<<<dim3(NT_ / 256, 128), 256, 0, stream>>>(bufA, dbA);
  k_agg<<<dim3(NT2_), 256, 0, stream>>>(bufA, wbA, bufB);

  // final linear (2048 -> 128) + BN over (B,S) -> d_out
  {
    const int nel = 128 * 2048;
    k_padw<<<dim3((nel + 255) / 256), 256, 0, stream>>>(lin_w, wpad, 128, 2048, 2048, nel);
    k_zero<<<1, 256, 0, stream>>>(stats, 256);
    k_gemm<2048, NT2_><<<dim3(NT2_ / 128, 8), 256, 0, stream>>>(wpad, bufB, ybuf);
    k_colstats<<<dim3(32, 128), 256, 0, stream>>>(ybuf, stats, NT2_);
    k_bn_out<<<dim3(NT2_ / 256, 128), 256, 0, stream>>>(ybuf, stats, bnl_g, bnl_b, out);
  }
  k_out_xyz<<<dim3(NT2_ / 256), 256, 0, stream>>>(nxyz, out);
}